// GCNConvNet_beta_22101901705658
// MI455X (gfx1250) — compile-verified
//
#include <hip/hip_runtime.h>
#include <hip/hip_bf16.h>

typedef float v2f __attribute__((ext_vector_type(2)));
typedef float v8f __attribute__((ext_vector_type(8)));

#define D 64

// ---------------- graph preprocessing ----------------

__global__ void k_zero_i32(int* __restrict__ p, int n) {
    int i = blockIdx.x * blockDim.x + threadIdx.x;
    if (i < n) p[i] = 0;
}

__global__ void k_count_deg(const int* __restrict__ dst, int* __restrict__ deg, int E) {
    int e = blockIdx.x * blockDim.x + threadIdx.x;
    if (e < E) atomicAdd(&deg[dst[e]], 1);
}

__global__ void k_isd(const int* __restrict__ deg, float* __restrict__ isd, int n) {
    int i = blockIdx.x * blockDim.x + threadIdx.x;
    if (i < n) isd[i] = rsqrtf((float)deg[i] + 1.0f);   // +1 for self-loop
}

// Single-block chunked exclusive scan (N = 50k is tiny); also seeds cursor.
__global__ __launch_bounds__(1024) void k_scan(const int* __restrict__ deg,
                                               int* __restrict__ row_ptr,
                                               int* __restrict__ cursor, int n) {
    __shared__ int sm[1024];
    __shared__ int carry_s;
    if (threadIdx.x == 0) carry_s = 0;
    __syncthreads();
    for (int base = 0; base < n; base += 1024) {
        int i = base + (int)threadIdx.x;
        int v = (i < n) ? deg[i] : 0;
        sm[threadIdx.x] = v;
        __syncthreads();
        for (int off = 1; off < 1024; off <<= 1) {
            int t = (threadIdx.x >= (unsigned)off) ? sm[threadIdx.x - off] : 0;
            __syncthreads();
            sm[threadIdx.x] += t;
            __syncthreads();
        }
        int incl = sm[threadIdx.x];
        int carry = carry_s;
        if (i < n) { int ex = carry + incl - v; row_ptr[i] = ex; cursor[i] = ex; }
        __syncthreads();
        if (threadIdx.x == 0) carry_s = carry + sm[1023];
        __syncthreads();
    }
    if (threadIdx.x == 0) row_ptr[n] = carry_s;
}

__global__ void k_fill(const int* __restrict__ src, const int* __restrict__ dst,
                       int* __restrict__ cursor, int* __restrict__ col, int E) {
    int e = blockIdx.x * blockDim.x + threadIdx.x;
    if (e < E) {
        int pos = atomicAdd(&cursor[dst[e]], 1);
        col[pos] = src[e];
    }
}

// Insertion-sort each row's source list so the fp summation order is
// deterministic across calls (atomic fill order is not).
__global__ void k_sort_rows(const int* __restrict__ row_ptr, int* __restrict__ col, int n) {
    int i = blockIdx.x * blockDim.x + threadIdx.x;
    if (i >= n) return;
    int s0 = row_ptr[i], s1 = row_ptr[i + 1];
    for (int a = s0 + 1; a < s1; a++) {
        int key = col[a];
        int j = a - 1;
        while (j >= s0 && col[j] > key) { col[j + 1] = col[j]; j--; }
        col[j + 1] = key;
    }
}

// ---------------- out = in @ W^T + b via V_WMMA_F32_16X16X4_F32 ----------------
// One wave32 computes a 16x64 output tile (4 accumulators of 16x16), K in steps of 4.

__global__ __launch_bounds__(128) void k_linear64_wmma(const float* __restrict__ in,
                                                       const float* __restrict__ W,
                                                       const float* __restrict__ b,
                                                       float* __restrict__ out, int n) {
    __shared__ __align__(16) float Ws[D * D];   // W[out][in], row-major (as given)
    __shared__ __align__(16) float bs[D];
    __shared__ __align__(16) float As[4][16 * D];

    int tid = threadIdx.x;
    for (int i = tid * 4; i < D * D; i += 128 * 4)
        *(float4*)&Ws[i] = *(const float4*)&W[i];
    if (tid < D) bs[tid] = b[tid];

    int wave = tid >> 5, lane = tid & 31;
    int tile = (blockIdx.x * 4 + wave) * 16;

    // Stage this wave's 16x64 A tile (rows clamped; compute garbage, store guarded).
    for (int i = lane; i < 16 * 16; i += 32) {
        int r = i >> 4, c4 = (i & 15) * 4;
        int row = tile + r;
        row = row < n ? row : (n - 1);
        *(float4*)&As[wave][r * D + c4] = *(const float4*)&in[(size_t)row * D + c4];
    }
    __syncthreads();
    if (tile >= n) return;   // uniform per-wave, after the only barrier

    int m = lane & 15, half = lane >> 4;

    v8f acc[4];
    #pragma unroll
    for (int nb = 0; nb < 4; nb++) {
        float bv = bs[nb * 16 + m];          // bias depends only on out-column
        #pragma unroll
        for (int i = 0; i < 8; i++) acc[nb][i] = bv;
    }

    // A frag (16x4, ISA layout): a[j] = X[m][k0 + 2*half + j]
    // B frag (4x16):             b[j] = W^T[2*half+j][m+16nb] = W[m+16nb][k0+2*half+j]
    #pragma unroll 4
    for (int k0 = 0; k0 < D; k0 += 4) {
        v2f a = *(const v2f*)&As[wave][m * D + k0 + 2 * half];
        #pragma unroll
        for (int nb = 0; nb < 4; nb++) {
            v2f bb = *(const v2f*)&Ws[(nb * 16 + m) * D + k0 + 2 * half];
            acc[nb] = __builtin_amdgcn_wmma_f32_16x16x4_f32(
                false, a, false, bb, (short)0, acc[nb], false, false);
        }
    }

    // C/D layout: VGPR i -> row (i + 8*half), col = m (+16*nb).
    // Fast path: whole 16-row tile in bounds (always true when n % 16 == 0,
    // which holds for N=50000) -> unguarded, coalesced b32 stores, no exec churn.
    float* outb = out + (size_t)(tile + half * 8) * D + m;
    if (tile + 16 <= n) {
        #pragma unroll
        for (int nb = 0; nb < 4; nb++) {
            #pragma unroll
            for (int i = 0; i < 8; i++)
                outb[(size_t)i * D + nb * 16] = acc[nb][i];
        }
    } else {
        #pragma unroll
        for (int nb = 0; nb < 4; nb++) {
            #pragma unroll
            for (int i = 0; i < 8; i++) {
                int row = tile + half * 8 + i;
                if (row < n) out[(size_t)row * D + nb * 16 + m] = acc[nb][i];
            }
        }
    }
}

// ---------------- aggregation: one wave per node, lane owns a float2 ----------------
// out[n] = isd[n] * sum_{s in N(n)} isd[s]*h[s] + h[n]*isd[n]^2   (optional ReLU)

__global__ __launch_bounds__(256) void k_agg64(const float* __restrict__ hlin,
                                               const int* __restrict__ row_ptr,
                                               const int* __restrict__ col,
                                               const float* __restrict__ isd,
                                               float* __restrict__ hout, int n, int relu) {
    int wid  = (blockIdx.x * blockDim.x + threadIdx.x) >> 5;
    int lane = threadIdx.x & 31;
    if (wid >= n) return;
    int s0 = row_ptr[wid], s1 = row_ptr[wid + 1];
    float dn = isd[wid];
    const float* hbase = hlin + lane * 2;
    float ax = 0.f, ay = 0.f;
    for (int e = s0; e < s1; e++) {
        int s = col[e];
        if (e + 1 < s1) {
            int s2 = col[e + 1];
            __builtin_prefetch(&hbase[(size_t)s2 * D], 0, 0);
        }
        float w = isd[s];
        float2 hv = *(const float2*)&hbase[(size_t)s * D];
        ax = fmaf(w, hv.x, ax);
        ay = fmaf(w, hv.y, ay);
    }
    float2 self = *(const float2*)&hbase[(size_t)wid * D];
    float sc = dn * dn;
    float ox = fmaf(dn, ax, self.x * sc);
    float oy = fmaf(dn, ay, self.y * sc);
    if (relu) { ox = fmaxf(ox, 0.f); oy = fmaxf(oy, 0.f); }
    *(float2*)&hout[(size_t)wid * D + lane * 2] = make_float2(ox, oy);
}

// ---------------- layer 2: [N,64] @ [64,1] + b ----------------

__global__ void k_dot64(const float* __restrict__ h, const float* __restrict__ W2,
                        const float* __restrict__ b2, float* __restrict__ vec, int n) {
    __shared__ float w[D];
    if (threadIdx.x < D) w[threadIdx.x] = W2[threadIdx.x];
    __syncthreads();
    int i = blockIdx.x * blockDim.x + threadIdx.x;
    if (i >= n) return;
    const float* row = h + (size_t)i * D;
    float acc = 0.f;
    #pragma unroll
    for (int k = 0; k < D; k += 4) {
        float4 hv = *(const float4*)&row[k];
        acc = fmaf(hv.x, w[k], acc);
        acc = fmaf(hv.y, w[k + 1], acc);
        acc = fmaf(hv.z, w[k + 2], acc);
        acc = fmaf(hv.w, w[k + 3], acc);
    }
    vec[i] = acc + b2[0];
}

__global__ void k_agg1(const float* __restrict__ vec, const int* __restrict__ row_ptr,
                       const int* __restrict__ col, const float* __restrict__ isd,
                       float* __restrict__ out, int n) {
    int i = blockIdx.x * blockDim.x + threadIdx.x;
    if (i >= n) return;
    int s0 = row_ptr[i], s1 = row_ptr[i + 1];
    float acc = 0.f;
    for (int e = s0; e < s1; e++) { int s = col[e]; acc = fmaf(isd[s], vec[s], acc); }
    float d = isd[i];
    out[i] = fmaf(d, acc, vec[i] * d * d);
}

// ---------------- host ----------------

extern "C" void kernel_launch(void* const* d_in, const int* in_sizes, int n_in,
                              void* d_out, int out_size, void* d_ws, size_t ws_size,
                              hipStream_t stream) {
    const float* x  = (const float*)d_in[0];
    const int*   ei = (const int*)d_in[1];   // jax x64 disabled -> int32 on device
    const float* W0 = (const float*)d_in[2];
    const float* b0 = (const float*)d_in[3];
    const float* W1 = (const float*)d_in[4];
    const float* b1 = (const float*)d_in[5];
    const float* W2 = (const float*)d_in[6];
    const float* b2 = (const float*)d_in[7];
    float* out = (float*)d_out;

    int n = in_sizes[0] / D;
    int E = in_sizes[1] / 2;
    const int* src = ei;
    const int* dst = ei + E;

    char* p = (char*)d_ws;
    auto alloc = [&](size_t bytes) -> void* {
        void* r = (void*)p;
        p += (bytes + 255) & ~(size_t)255;
        return r;
    };
    int*   deg     = (int*)alloc((size_t)n * sizeof(int));
    int*   row_ptr = (int*)alloc((size_t)(n + 1) * sizeof(int));
    int*   cursor  = (int*)alloc((size_t)n * sizeof(int));
    int*   col     = (int*)alloc((size_t)E * sizeof(int));
    float* isd     = (float*)alloc((size_t)n * sizeof(float));
    float* lin     = (float*)alloc((size_t)n * D * sizeof(float));
    float* hbuf    = (float*)alloc((size_t)n * D * sizeof(float));
    float* vec     = (float*)alloc((size_t)n * sizeof(float));
    (void)ws_size; (void)n_in; (void)out_size;

    const int tb = 256;
    // Build degree-normalization + CSR-by-dst (reused by all 3 layers)
    k_zero_i32 <<<(n + tb - 1) / tb, tb, 0, stream>>>(deg, n);
    k_count_deg<<<(E + tb - 1) / tb, tb, 0, stream>>>(dst, deg, E);
    k_isd      <<<(n + tb - 1) / tb, tb, 0, stream>>>(deg, isd, n);
    k_scan     <<<1, 1024, 0, stream>>>(deg, row_ptr, cursor, n);
    k_fill     <<<(E + tb - 1) / tb, tb, 0, stream>>>(src, dst, cursor, col, E);
    k_sort_rows<<<(n + tb - 1) / tb, tb, 0, stream>>>(row_ptr, col, n);

    int gblocks = (n + 63) / 64;            // 128 threads = 4 waves, 16 rows each
    int ablocks = ((n * 32) + tb - 1) / tb; // one wave per node

    // layer 0
    k_linear64_wmma<<<gblocks, 128, 0, stream>>>(x, W0, b0, lin, n);
    k_agg64<<<ablocks, tb, 0, stream>>>(lin, row_ptr, col, isd, hbuf, n, 1);
    // layer 1
    k_linear64_wmma<<<gblocks, 128, 0, stream>>>(hbuf, W1, b1, lin, n);
    k_agg64<<<ablocks, tb, 0, stream>>>(lin, row_ptr, col, isd, hbuf, n, 1);
    // layer 2
    k_dot64<<<(n + tb - 1) / tb, tb, 0, stream>>>(hbuf, W2, b2, vec, n);
    k_agg1 <<<(n + tb - 1) / tb, tb, 0, stream>>>(vec, row_ptr, col, isd, out, n);
}